// DetectionLoss_61409442398458
// MI455X (gfx1250) — compile-verified
//
#include <hip/hip_runtime.h>
#include <hip/hip_bf16.h>
#include <stdint.h>

// ---------------- CDNA5 vector types ----------------
typedef __attribute__((ext_vector_type(2))) float v2f;
typedef __attribute__((ext_vector_type(4))) float v4f;
typedef __attribute__((ext_vector_type(8))) float v8f;

#define EPS_ 1e-7f
#define NA 3
#define NC 20
#define CHT 75          // NA*(5+NC) channels
#define OBJ_BLOCKS 512
#define TPB 256
#define FOUR_OVER_PI2 0.4052847345693511f

__device__ __forceinline__ float softplusf_(float x) {
    // log(1+e^x), numerically stable; BCE(l,y) = softplus(l) - y*l
    return fmaxf(x, 0.f) + log1pf(expf(-fabsf(x)));
}
__device__ __forceinline__ float sigm_(float x) { return 1.f / (1.f + expf(-x)); }

// ---------------- zero the target mask ----------------
__global__ void k_zero(uint32_t* __restrict__ w, int n) {
    int i = blockIdx.x * blockDim.x + threadIdx.x;
    int stride = gridDim.x * blockDim.x;
    for (; i < n; i += stride) w[i] = 0u;
}

// ---------------- scatter targets into per-(b,h,w) mask (dedup via idempotent store) ----------------
__global__ void k_scatter(const float* __restrict__ t, uint8_t* __restrict__ mask,
                          int N, int B) {
    int n = blockIdx.x * blockDim.x + threadIdx.x;
    if (n >= N) return;
    int bi = (int)t[n * 6 + 0];
    float cxn = t[n * 6 + 2], cyn = t[n * 6 + 3];
    int off = 0;
    const int Ws[3] = {80, 40, 20};
    for (int s = 0; s < 3; ++s) {
        int W = Ws[s], HW = W * W;
        float cx = cxn * (float)W, cy = cyn * (float)W;   // H == W per scale
        int gi = (int)fminf(fmaxf(cx, 0.f), (float)(W - 1));
        int gj = (int)fminf(fmaxf(cy, 0.f), (float)(W - 1));
        mask[off + bi * HW + gj * W + gi] = 1;
        off += B * HW;
    }
}

// ---------------- big masked obj-BCE reduction over all 3 scales ----------------
// Vectorized: b128 logit loads + u32 mask loads. HW % 4 == 0 for every scale and
// hw is the fastest index, so a float4 chunk never crosses a (b,a)-plane boundary.
__global__ void k_obj(const float* __restrict__ p0, const float* __restrict__ p1,
                      const float* __restrict__ p2, const uint8_t* __restrict__ mask,
                      float* __restrict__ partials, int B) {
    const int e0 = B * NA * 6400, e1 = B * NA * 1600, e2 = B * NA * 400;
    const int q0 = e0 >> 2, q1 = e1 >> 2, q2 = e2 >> 2;   // vec4 counts
    const int EQ = q0 + q1 + q2;
    const float w0 = 1.f / (float)e0, w1 = 1.f / (float)e1, w2 = 1.f / (float)e2;
    int v = blockIdx.x * blockDim.x + threadIdx.x;
    int stride = gridDim.x * blockDim.x;
    float acc = 0.f;
    for (; v < EQ; v += stride) {
        const float* p; int HW, moff, remq; float wn;
        if (v < q0)           { p = p0; HW = 6400; moff = 0;                   remq = v;           wn = w0; }
        else if (v < q0 + q1) { p = p1; HW = 1600; moff = B * 6400;            remq = v - q0;      wn = w1; }
        else                  { p = p2; HW = 400;  moff = B * 6400 + B * 1600; remq = v - q0 - q1; wn = w2; }
        int i4 = remq << 2;                       // first element of this float4 chunk
        int b  = i4 / (NA * HW);
        int r  = i4 - b * NA * HW;
        int a  = r / HW;
        int hw = r - a * HW;                      // multiple of 4
        v4f l4 = *(const v4f*)(p + (size_t)(b * CHT + a * 25 + 4) * (size_t)HW + hw);
        uint32_t m4 = *(const uint32_t*)(mask + moff + b * HW + hw);
        float part = 0.f;
#pragma unroll
        for (int k = 0; k < 4; ++k) {
            float l = l4[k];
            float bce = softplusf_(l);                       // BCE(l,0)
            if ((m4 >> (8 * k)) & 0xFFu) bce -= l;           // BCE(l,1) = BCE(l,0) - l
            part += bce;
        }
        acc += wn * part;
    }
    __shared__ float red[TPB];
    red[threadIdx.x] = acc;
    __syncthreads();
    for (int s = TPB / 2; s > 0; s >>= 1) {
        if (threadIdx.x < s) red[threadIdx.x] += red[threadIdx.x + s];
        __syncthreads();
    }
    if (threadIdx.x == 0) partials[blockIdx.x] = red[0];
}

// ---------------- per-target CIoU box loss + class BCE (gather) ----------------
__global__ void k_boxcls(const float* __restrict__ p0, const float* __restrict__ p1,
                         const float* __restrict__ p2, const float* __restrict__ t,
                         float* __restrict__ boxP, float* __restrict__ clsP, int N) {
    int s = blockIdx.x;                               // 0..2 = scale
    const float* p = (s == 0) ? p0 : (s == 1) ? p1 : p2;
    int W = (s == 0) ? 80 : (s == 1) ? 40 : 20;
    int HW = W * W;
    float invW = 1.f / (float)W;
    float boxA = 0.f, clsA = 0.f;
    for (int n = threadIdx.x; n < N; n += TPB) {
        float t0 = t[n*6+0], t1 = t[n*6+1], t2 = t[n*6+2];
        float t3 = t[n*6+3], t4 = t[n*6+4], t5 = t[n*6+5];
        int bi = (int)t0, ci = (int)t1;
        float cx = t2 * (float)W, cy = t3 * (float)W;
        float tw = t4 * (float)W, th = t5 * (float)W;
        int gi = (int)fminf(fmaxf(cx, 0.f), (float)(W - 1));
        int gj = (int)fminf(fmaxf(cy, 0.f), (float)(W - 1));
        float gif = (float)gi, gjf = (float)gj;
        float tx1 = t2 - 0.5f * t4, ty1 = t3 - 0.5f * t5;
        float tx2 = t2 + 0.5f * t4, ty2 = t3 + 0.5f * t5;
        for (int a = 0; a < NA; ++a) {
            size_t idx0 = (size_t)(bi * CHT + a * 25) * (size_t)HW + gj * W + gi;
            float sx = sigm_(p[idx0]);
            float sy = sigm_(p[idx0 + HW]);
            float px1 = (sx + gif - 0.5f * tw) * invW;
            float py1 = (sy + gjf - 0.5f * th) * invW;
            float px2 = (sx + gif + 0.5f * tw) * invW;
            float py2 = (sy + gjf + 0.5f * th) * invW;
            // CIoU
            float ix1 = fmaxf(px1, tx1), iy1 = fmaxf(py1, ty1);
            float ix2 = fminf(px2, tx2), iy2 = fminf(py2, ty2);
            float inter = fmaxf(ix2 - ix1, 0.f) * fmaxf(iy2 - iy1, 0.f);
            float areaP = (px2 - px1) * (py2 - py1);
            float areaT = (tx2 - tx1) * (ty2 - ty1);
            float uni = areaP + areaT - inter + EPS_;
            float iou = inter / uni;
            float ex1 = fminf(px1, tx1), ey1 = fminf(py1, ty1);
            float ex2 = fmaxf(px2, tx2), ey2 = fmaxf(py2, ty2);
            float c2 = (ex2 - ex1) * (ex2 - ex1) + (ey2 - ey1) * (ey2 - ey1) + EPS_;
            float dcx = 0.5f * (px1 + px2) - 0.5f * (tx1 + tx2);
            float dcy = 0.5f * (py1 + py2) - 0.5f * (ty1 + ty2);
            float rho2 = dcx * dcx + dcy * dcy;
            float pw  = fmaxf(px2 - px1, EPS_), ph  = fmaxf(py2 - py1, EPS_);
            float twc = fmaxf(tx2 - tx1, EPS_), thc = fmaxf(ty2 - ty1, EPS_);
            float dv = atanf(twc / thc) - atanf(pw / ph);
            float v = FOUR_OVER_PI2 * dv * dv;
            float alpha = v / (1.f - iou + v + EPS_);
            float ciou = iou - rho2 / c2 - alpha * v;
            boxA += 1.f - ciou;
            // class BCE, mean over NC:  sum_c softplus(l_c) - l_ci
            float lsum = 0.f;
            for (int c = 0; c < NC; ++c) {
                float l = p[idx0 + (size_t)(5 + c) * (size_t)HW];
                lsum += softplusf_(l);
                if (c == ci) lsum -= l;
            }
            clsA += lsum * (1.f / (float)NC);
        }
    }
    __shared__ float redB[TPB];
    __shared__ float redC[TPB];
    redB[threadIdx.x] = boxA;
    redC[threadIdx.x] = clsA;
    __syncthreads();
    for (int st = TPB / 2; st > 0; st >>= 1) {
        if (threadIdx.x < st) {
            redB[threadIdx.x] += redB[threadIdx.x + st];
            redC[threadIdx.x] += redC[threadIdx.x + st];
        }
        __syncthreads();
    }
    if (threadIdx.x == 0) { boxP[s] = redB[0]; clsP[s] = redC[0]; }
}

// ---------------- finalize: wave32 WMMA ones-matrix reduction + combine ----------------
// Layout-agnostic identity: with B = all-ones and C = 0,
//   sum_{m,n} D[m][n] = 16 * sum(all A entries)   (f32 matrix pipe, no precision loss)
__global__ void k_final(const float* __restrict__ objP, int nObj,
                        const float* __restrict__ boxP, const float* __restrict__ clsP,
                        float* __restrict__ out, float invNumT) {
    int lane = threadIdx.x;          // exactly 32 threads; EXEC all-ones for WMMA
    float ax = 0.f;
    for (int i = lane; i < nObj; i += 32) ax += objP[i];
    v2f a; a[0] = ax; a[1] = 0.f;    // each lane contributes its partial into A
    v2f b; b[0] = 1.f; b[1] = 1.f;   // B = ones
    v8f c = {};
    v8f d = __builtin_amdgcn_wmma_f32_16x16x4_f32(
        /*neg_a=*/false, a, /*neg_b=*/false, b,
        /*c_mod=*/(short)0, c, /*reuse_a=*/false, /*reuse_b=*/false);
    float s = d[0] + d[1] + d[2] + d[3] + d[4] + d[5] + d[6] + d[7];
    __shared__ float red[32];
    red[lane] = s;
    __syncthreads();
    if (lane == 0) {
        float tot = 0.f;
        for (int i = 0; i < 32; ++i) tot += red[i];   // fixed order -> deterministic
        float lo = tot * (1.f / 16.f);                // undo the 16x fan-out of ones-B
        float lb = (boxP[0] + boxP[1] + boxP[2]) * invNumT;
        float lc = (clsP[0] + clsP[1] + clsP[2]) * invNumT;
        float total = 0.05f * lb + 1.0f * lo + 0.5f * lc;
        out[0] = total; out[1] = lb; out[2] = lo; out[3] = lc; out[4] = 0.f;
    }
}

extern "C" void kernel_launch(void* const* d_in, const int* in_sizes, int n_in,
                              void* d_out, int out_size, void* d_ws, size_t ws_size,
                              hipStream_t stream) {
    const float* p0 = (const float*)d_in[0];
    const float* p1 = (const float*)d_in[1];
    const float* p2 = (const float*)d_in[2];
    const float* tg = (const float*)d_in[3];
    float* out = (float*)d_out;

    int N = in_sizes[3] / 6;                 // number of targets (256)
    int B = in_sizes[0] / (CHT * 6400);      // batch (32)

    // workspace layout: [mask bytes][obj partials][box partials][cls partials]
    size_t maskBytes = (size_t)B * 8400u;    // B*(80^2+40^2+20^2)
    size_t maskAl = (maskBytes + 255) & ~(size_t)255;
    uint8_t* mask = (uint8_t*)d_ws;
    float* objP = (float*)((char*)d_ws + maskAl);
    float* boxP = objP + OBJ_BLOCKS;
    float* clsP = boxP + 4;

    int words = (int)(maskBytes / 4);
    k_zero<<<(words + TPB - 1) / TPB, TPB, 0, stream>>>((uint32_t*)mask, words);
    k_scatter<<<(N + TPB - 1) / TPB, TPB, 0, stream>>>(tg, mask, N, B);
    k_obj<<<OBJ_BLOCKS, TPB, 0, stream>>>(p0, p1, p2, mask, objP, B);
    k_boxcls<<<3, TPB, 0, stream>>>(p0, p1, p2, tg, boxP, clsP, N);

    int numT = N * NA * 3; if (numT < 1) numT = 1;
    float invNumT = 1.f / (float)numT;
    k_final<<<1, 32, 0, stream>>>(objP, OBJ_BLOCKS, boxP, clsP, out, invNumT);
}